// S4Model_25761213841677
// MI455X (gfx1250) — compile-verified
//
#include <hip/hip_runtime.h>
#include <hip/hip_bf16.h>

#define BB 16
#define DIN 16
#define LL 4096
#define HH 256
#define NN 32
#define NLAYER 4
#define CT 128            // chunk length in time
#define NC 32             // LL / CT chunks
#define KW 192            // CT + 2*NN : width of [M | V]

typedef __attribute__((ext_vector_type(2))) float v2f;
typedef __attribute__((ext_vector_type(8))) float v8f;

static __device__ __forceinline__ v8f wmma4(v2f a, v2f b, v8f c) {
  // V_WMMA_F32_16X16X4_F32 : D = A(16x4) * B(4x16) + C  (fp32 exact)
  return __builtin_amdgcn_wmma_f32_16x16x4_f32(false, a, false, b, (short)0, c, false, false);
}

static __device__ __forceinline__ v8f zero8() {
  v8f z = {0.f, 0.f, 0.f, 0.f, 0.f, 0.f, 0.f, 0.f};
  return z;
}

static __device__ __forceinline__ float gelu_tanh(float y) {
  float y3 = y * y * y;
  float t = tanhf(0.7978845608028654f * (y + 0.044715f * y3));
  return 0.5f * y * (1.f + t);
}

// ---------------------------------------------------------------------------
// Prep 1: per (h,n) mode -> coeff (w, 2*Cc), E rows (w^(T-1-t)), V cols of MV
// ---------------------------------------------------------------------------
__global__ void prep_coeff_ev(const float* __restrict__ log_dt,
                              const float* __restrict__ A_re,
                              const float* __restrict__ A_im,
                              const float* __restrict__ C_re,
                              const float* __restrict__ C_im,
                              float4* __restrict__ coeff,
                              float* __restrict__ Ebuf,
                              float* __restrict__ MVbuf) {
  int idx = blockIdx.x * blockDim.x + threadIdx.x;   // HH*NN
  if (idx >= HH * NN) return;
  int h = idx / NN, n = idx % NN;
  float dt = __expf(log_dt[h]);
  float ar = -__expf(A_re[idx]);
  float ai = A_im[idx];
  float dr = ar * dt, di = ai * dt;
  float es = __expf(dr);
  float sn, cs;
  __sincosf(di, &sn, &cs);
  float wr = es * cs, wi = es * sn;                  // w = exp(dt*A)
  float inv = 1.0f / (ar * ar + ai * ai);
  float numr = wr - 1.f, numi = wi;                  // (w - 1)/A
  float qr = (numr * ar + numi * ai) * inv;
  float qi = (numi * ar - numr * ai) * inv;
  float cr = C_re[idx], ci = C_im[idx];
  float c2r = 2.f * (cr * qr - ci * qi);             // 2*Cc (discretized)
  float c2i = 2.f * (cr * qi + ci * qr);
  coeff[n * HH + h] = make_float4(wr, wi, c2r, c2i);

  float* E0 = Ebuf + ((size_t)h * 64 + 2 * n) * CT;
  float* E1 = E0 + CT;
  float qre = 1.f, qim = 0.f;                        // w^t
  float pre = c2r * wr - c2i * wi;                   // 2Cc * w^(t+1)
  float pim = c2r * wi + c2i * wr;
  for (int t = 0; t < CT; ++t) {
    E0[CT - 1 - t] = qre;                            // E[2n][t'] = Re(w^(T-1-t'))
    E1[CT - 1 - t] = qim;
    float* mv = MVbuf + ((size_t)h * CT + t) * KW;
    mv[CT + 2 * n]     = pre;                        // V[t][2n]   =  Re(2Cc w^(t+1))
    mv[CT + 2 * n + 1] = -pim;                       // V[t][2n+1] = -Im(2Cc w^(t+1))
    float nq = qre * wr - qim * wi; qim = qre * wi + qim * wr; qre = nq;
    float np = pre * wr - pim * wi; pim = pre * wi + pim * wr; pre = np;
  }
}

// ---------------------------------------------------------------------------
// Prep 2: K[h][t] = Re( sum_n 2Cc_n * w_n^t )
// ---------------------------------------------------------------------------
__global__ void prep_K(const float4* __restrict__ coeff,
                       const float* __restrict__ log_dt,
                       const float* __restrict__ A_re,
                       const float* __restrict__ A_im,
                       float* __restrict__ Kbuf) {
  int idx = blockIdx.x * blockDim.x + threadIdx.x;   // HH*CT
  if (idx >= HH * CT) return;
  int h = idx / CT, t = idx % CT;
  float dt = __expf(log_dt[h]);
  float acc = 0.f;
  for (int n = 0; n < NN; ++n) {
    float dr = -__expf(A_re[h * NN + n]) * dt;
    float di = A_im[h * NN + n] * dt;
    float e = __expf(dr * (float)t);
    float sn, cs;
    __sincosf(di * (float)t, &sn, &cs);
    float4 c = coeff[n * HH + h];
    acc += c.z * (e * cs) - c.w * (e * sn);
  }
  Kbuf[h * CT + t] = acc;
}

// ---------------------------------------------------------------------------
// Prep 3: lower-triangular Toeplitz M[t][s] = K[t-s]; D-skip folded on diagonal
// ---------------------------------------------------------------------------
__global__ void prep_Mfill(const float* __restrict__ Kbuf,
                           const float* __restrict__ Dvec,
                           float* __restrict__ MVbuf) {
  int idx = blockIdx.x * blockDim.x + threadIdx.x;   // HH*CT
  if (idx >= HH * CT) return;
  int h = idx / CT, t = idx % CT;
  const float* Kr = Kbuf + h * CT;
  float* mv = MVbuf + ((size_t)h * CT + t) * KW;
  float dskip = Dvec[h];
  for (int s = 0; s < CT; ++s) {
    float v = 0.f;
    if (s < t) v = Kr[t - s];
    else if (s == t) v = Kr[0] + dskip;
    mv[s] = v;
  }
}

// ---------------------------------------------------------------------------
// GEMM 1 (WMMA): per h, S(64 x 512) = E_h(64 x 128) * Z(128 x 512)
// cols = (b, chunk). Stores chunk-local end states.
// ---------------------------------------------------------------------------
__global__ void __launch_bounds__(256) gemm_states(const float* __restrict__ Ebuf,
                                                   const float* __restrict__ z,
                                                   float* __restrict__ states) {
  int h = blockIdx.x >> 2;
  int mtile = blockIdx.x & 3;
  int wave = threadIdx.x >> 5, lane = threadIdx.x & 31;
  int nlo = lane & 15, kh = lane >> 4;
  const float* Eh = Ebuf + (size_t)h * 64 * CT;
  int m = mtile * 16 + nlo;
  v8f acc[4] = {zero8(), zero8(), zero8(), zero8()};
  for (int k = 0; k < CT; k += 4) {
    int kk = k + 2 * kh;
    v2f a;
    a.x = Eh[m * CT + kk];
    a.y = Eh[m * CT + kk + 1];
#pragma unroll
    for (int j = 0; j < 4; ++j) {
      int col = (wave * 4 + j) * 16 + nlo;           // 0..511
      int b = col >> 5, c = col & 31;
      const float* zp = z + ((size_t)(b * HH + h)) * LL + c * CT + kk;
      v2f bb;
      bb.x = zp[0];
      bb.y = zp[1];
      acc[j] = wmma4(a, bb, acc[j]);
    }
  }
#pragma unroll
  for (int j = 0; j < 4; ++j) {
    int col = (wave * 4 + j) * 16 + nlo;
    int b = col >> 5, c = col & 31;
    float* sp = states + (((size_t)(b * HH + h)) * NC + c) * 64 + mtile * 16 + kh * 8;
    *(float4*)sp = make_float4(acc[j][0], acc[j][1], acc[j][2], acc[j][3]);
    *(float4*)(sp + 4) = make_float4(acc[j][4], acc[j][5], acc[j][6], acc[j][7]);
  }
}

// ---------------------------------------------------------------------------
// Combine: 32-step diagonal prefix over chunks (s-space), in place:
//   states[c] <- End_{c-1};  End_c = S_c + w^CT * End_{c-1}
// ---------------------------------------------------------------------------
__global__ void combine_kernel(const float4* __restrict__ coeff,
                               float* __restrict__ states) {
  int idx = blockIdx.x * blockDim.x + threadIdx.x;   // BB*HH*NN
  if (idx >= BB * HH * NN) return;
  int n = idx & (NN - 1);
  int bh = idx / NN;
  int h = bh & (HH - 1);
  float4 cw = coeff[n * HH + h];
  float wr = cw.x, wi = cw.y;
#pragma unroll
  for (int i = 0; i < 7; ++i) {                      // w^128 by squaring
    float nr = wr * wr - wi * wi;
    wi = 2.f * wr * wi;
    wr = nr;
  }
  float er = 0.f, ei = 0.f;
  float* sp = states + (size_t)bh * NC * 64 + 2 * n;
#pragma unroll
  for (int c = 0; c < NC; ++c) {
    float sr = sp[0], si = sp[1];
    sp[0] = er;
    sp[1] = ei;
    float tr = wr * er - wi * ei + sr;
    ei = wr * ei + wi * er + si;
    er = tr;
    sp += 64;
  }
}

// ---------------------------------------------------------------------------
// GEMM 2 (WMMA): per h, Y(128 x 512) = [M|V](128 x 192) * [Z ; init](192 x 512)
// fused GELU, writes g.  (skip term already folded into M's diagonal)
// ---------------------------------------------------------------------------
__global__ void __launch_bounds__(256) gemm_y(const float* __restrict__ MVbuf,
                                              const float* __restrict__ z,
                                              const float* __restrict__ states,
                                              float* __restrict__ g) {
  int h = blockIdx.x >> 3;
  int mtile = blockIdx.x & 7;
  int wave = threadIdx.x >> 5, lane = threadIdx.x & 31;
  int nlo = lane & 15, kh = lane >> 4;
  const float* MVh = MVbuf + (size_t)h * CT * KW;
  int m = mtile * 16 + nlo;
  v8f acc[4] = {zero8(), zero8(), zero8(), zero8()};
  for (int k = 0; k < CT; k += 4) {                  // conv part: B rows from z
    int kk = k + 2 * kh;
    v2f a;
    a.x = MVh[m * KW + kk];
    a.y = MVh[m * KW + kk + 1];
    __builtin_prefetch(MVh + m * KW + kk + 32, 0, 1);
#pragma unroll
    for (int j = 0; j < 4; ++j) {
      int col = (wave * 4 + j) * 16 + nlo;
      int b = col >> 5, c = col & 31;
      const float* zp = z + ((size_t)(b * HH + h)) * LL + c * CT + kk;
      v2f bb;
      bb.x = zp[0];
      bb.y = zp[1];
      acc[j] = wmma4(a, bb, acc[j]);
    }
  }
  for (int k = CT; k < KW; k += 4) {                 // correction: B rows from init states
    int kk = k + 2 * kh;
    v2f a;
    a.x = MVh[m * KW + kk];
    a.y = MVh[m * KW + kk + 1];
#pragma unroll
    for (int j = 0; j < 4; ++j) {
      int col = (wave * 4 + j) * 16 + nlo;
      int b = col >> 5, c = col & 31;
      const float* sp = states + (((size_t)(b * HH + h)) * NC + c) * 64 + (kk - CT);
      v2f bb;
      bb.x = sp[0];
      bb.y = sp[1];
      acc[j] = wmma4(a, bb, acc[j]);
    }
  }
#pragma unroll
  for (int j = 0; j < 4; ++j) {
    int col = (wave * 4 + j) * 16 + nlo;
    int b = col >> 5, c = col & 31;
    float* gp = g + ((size_t)(b * HH + h)) * LL + c * CT + mtile * 16 + kh * 8;
    *(float4*)gp = make_float4(gelu_tanh(acc[j][0]), gelu_tanh(acc[j][1]),
                               gelu_tanh(acc[j][2]), gelu_tanh(acc[j][3]));
    *(float4*)(gp + 4) = make_float4(gelu_tanh(acc[j][4]), gelu_tanh(acc[j][5]),
                                     gelu_tanh(acc[j][6]), gelu_tanh(acc[j][7]));
  }
}

// ---------------------------------------------------------------------------
// Encoder (WMMA): h[b,d,l] = sum_i x[b,i,l] * We[i,d] + be[d]
// ---------------------------------------------------------------------------
__global__ void __launch_bounds__(256) encoder_kernel(const float* __restrict__ x,
                                                      const float* __restrict__ We,
                                                      const float* __restrict__ be,
                                                      float* __restrict__ hbuf) {
  int b = blockIdx.x >> 8;
  int l0 = (blockIdx.x & 255) * 16;
  int wave = threadIdx.x >> 5, lane = threadIdx.x & 31;
  int nlo = lane & 15, kh = lane >> 4;
  int d0 = wave * 32;
  v8f acc0 = zero8(), acc1 = zero8();
  const float* xb = x + (size_t)b * DIN * LL + l0 + nlo;
  for (int k = 0; k < DIN; k += 4) {
    int kk = k + 2 * kh;
    v2f a;
    a.x = xb[(size_t)kk * LL];
    a.y = xb[(size_t)(kk + 1) * LL];
    v2f b0, b1;
    b0.x = We[kk * HH + d0 + nlo];
    b0.y = We[(kk + 1) * HH + d0 + nlo];
    b1.x = We[kk * HH + d0 + 16 + nlo];
    b1.y = We[(kk + 1) * HH + d0 + 16 + nlo];
    acc0 = wmma4(a, b0, acc0);
    acc1 = wmma4(a, b1, acc1);
  }
  float bias0 = be[d0 + nlo], bias1 = be[d0 + 16 + nlo];
  float* o0 = hbuf + ((size_t)b * HH + d0 + nlo) * LL + l0 + kh * 8;
  float* o1 = hbuf + ((size_t)b * HH + d0 + 16 + nlo) * LL + l0 + kh * 8;
  *(float4*)o0 = make_float4(acc0[0] + bias0, acc0[1] + bias0, acc0[2] + bias0, acc0[3] + bias0);
  *(float4*)(o0 + 4) = make_float4(acc0[4] + bias0, acc0[5] + bias0, acc0[6] + bias0, acc0[7] + bias0);
  *(float4*)o1 = make_float4(acc1[0] + bias1, acc1[1] + bias1, acc1[2] + bias1, acc1[3] + bias1);
  *(float4*)(o1 + 4) = make_float4(acc1[4] + bias1, acc1[5] + bias1, acc1[6] + bias1, acc1[7] + bias1);
}

// ---------------------------------------------------------------------------
// W_out GEMM (WMMA) + bias + residual + LayerNorm over H, h updated in place.
// Block owns full (16 l x 256 d) stripe -> LN reduction in LDS.
// ---------------------------------------------------------------------------
__global__ void __launch_bounds__(256) wout_ln_kernel(const float* __restrict__ g,
                                                      const float* __restrict__ W,
                                                      const float* __restrict__ bias,
                                                      float* __restrict__ hbuf,
                                                      const float* __restrict__ lnw,
                                                      const float* __restrict__ lnb) {
  __shared__ float smem[HH * 16];     // [d][l] staging (16 KB)
  __shared__ float redS[256];
  __shared__ float redQ[256];
  __shared__ float statM[16];
  __shared__ float statR[16];
  int b = blockIdx.x >> 8;
  int l0 = (blockIdx.x & 255) * 16;
  int wave = threadIdx.x >> 5, lane = threadIdx.x & 31;
  int nlo = lane & 15, kh = lane >> 4;
  int d0 = wave * 32;
  v8f acc0 = zero8(), acc1 = zero8();
  const float* gb = g + (size_t)b * HH * LL + l0 + nlo;
  for (int k = 0; k < HH; k += 4) {
    int kk = k + 2 * kh;
    v2f a;
    a.x = gb[(size_t)kk * LL];
    a.y = gb[(size_t)(kk + 1) * LL];
    v2f b0, b1;
    b0.x = W[kk * HH + d0 + nlo];
    b0.y = W[(kk + 1) * HH + d0 + nlo];
    b1.x = W[kk * HH + d0 + 16 + nlo];
    b1.y = W[(kk + 1) * HH + d0 + 16 + nlo];
    acc0 = wmma4(a, b0, acc0);
    acc1 = wmma4(a, b1, acc1);
  }
  float bias0 = bias[d0 + nlo], bias1 = bias[d0 + 16 + nlo];
#pragma unroll
  for (int r = 0; r < 8; ++r) {
    smem[(d0 + nlo) * 16 + kh * 8 + r] = acc0[r] + bias0;
    smem[(d0 + 16 + nlo) * 16 + kh * 8 + r] = acc1[r] + bias1;
  }
  __syncthreads();

  int d = threadIdx.x;
  float* hrow = hbuf + ((size_t)b * HH + d) * LL + l0;
  float4 h0 = *(const float4*)(hrow);
  float4 h1 = *(const float4*)(hrow + 4);
  float4 h2 = *(const float4*)(hrow + 8);
  float4 h3 = *(const float4*)(hrow + 12);
  float hv[16] = {h0.x, h0.y, h0.z, h0.w, h1.x, h1.y, h1.z, h1.w,
                  h2.x, h2.y, h2.z, h2.w, h3.x, h3.y, h3.z, h3.w};
  float* sd = smem + d * 16;
#pragma unroll
  for (int t = 0; t < 16; ++t) sd[t] += hv[t];       // residual
  __syncthreads();

  int lidx = threadIdx.x & 15, grp = threadIdx.x >> 4;
  float s = 0.f, q = 0.f;
#pragma unroll
  for (int j = 0; j < 16; ++j) {
    float v = smem[(grp + 16 * j) * 16 + lidx];
    s += v;
    q += v * v;
  }
  redS[lidx * 16 + grp] = s;
  redQ[lidx * 16 + grp] = q;
  __syncthreads();
  if (threadIdx.x < 16) {
    float ss = 0.f, qq = 0.f;
    for (int j = 0; j < 16; ++j) {
      ss += redS[threadIdx.x * 16 + j];
      qq += redQ[threadIdx.x * 16 + j];
    }
    float mu = ss * (1.f / HH);
    float var = qq * (1.f / HH) - mu * mu;
    statM[threadIdx.x] = mu;
    statR[threadIdx.x] = rsqrtf(var + 1e-5f);
  }
  __syncthreads();
  float lw = lnw[d], lb2 = lnb[d];
#pragma unroll
  for (int t0 = 0; t0 < 16; t0 += 4) {
    float4 o;
    o.x = (sd[t0 + 0] - statM[t0 + 0]) * statR[t0 + 0] * lw + lb2;
    o.y = (sd[t0 + 1] - statM[t0 + 1]) * statR[t0 + 1] * lw + lb2;
    o.z = (sd[t0 + 2] - statM[t0 + 2]) * statR[t0 + 2] * lw + lb2;
    o.w = (sd[t0 + 3] - statM[t0 + 3]) * statR[t0 + 3] * lw + lb2;
    *(float4*)(hrow + t0) = o;
  }
}

// ---------------------------------------------------------------------------
// Decoder (WMMA): out[b,i,l] = sum_d h[b,d,l] * Wd[d,i] + bd[i]   (N = 16)
// ---------------------------------------------------------------------------
__global__ void __launch_bounds__(256) decoder_kernel(const float* __restrict__ hbuf,
                                                      const float* __restrict__ Wd,
                                                      const float* __restrict__ bd,
                                                      float* __restrict__ out) {
  int wtile = blockIdx.x * 8 + (threadIdx.x >> 5);   // 4096 tiles over (b,l)
  int b = wtile >> 8;
  int l0 = (wtile & 255) * 16;
  int lane = threadIdx.x & 31;
  int nlo = lane & 15, kh = lane >> 4;
  v8f acc = zero8();
  const float* hb = hbuf + (size_t)b * HH * LL + l0 + nlo;
  for (int k = 0; k < HH; k += 4) {
    int kk = k + 2 * kh;
    v2f a;
    a.x = hb[(size_t)kk * LL];
    a.y = hb[(size_t)(kk + 1) * LL];
    v2f bb;
    bb.x = Wd[kk * DIN + nlo];
    bb.y = Wd[(kk + 1) * DIN + nlo];
    acc = wmma4(a, bb, acc);
  }
  float bias = bd[nlo];
  float* o = out + ((size_t)b * DIN + nlo) * LL + l0 + kh * 8;
  *(float4*)o = make_float4(acc[0] + bias, acc[1] + bias, acc[2] + bias, acc[3] + bias);
  *(float4*)(o + 4) = make_float4(acc[4] + bias, acc[5] + bias, acc[6] + bias, acc[7] + bias);
}

// ---------------------------------------------------------------------------
extern "C" void kernel_launch(void* const* d_in, const int* in_sizes, int n_in,
                              void* d_out, int out_size, void* d_ws, size_t ws_size,
                              hipStream_t stream) {
  (void)in_sizes; (void)n_in; (void)out_size; (void)ws_size;
  const float* x      = (const float*)d_in[0];
  const float* enc_w  = (const float*)d_in[1];
  const float* enc_b  = (const float*)d_in[2];
  const float* log_dt = (const float*)d_in[3];
  const float* A_re   = (const float*)d_in[4];
  const float* A_im   = (const float*)d_in[5];
  const float* C_re   = (const float*)d_in[6];
  const float* C_im   = (const float*)d_in[7];
  const float* Dv     = (const float*)d_in[8];
  const float* W_out  = (const float*)d_in[9];
  const float* b_out  = (const float*)d_in[10];
  const float* ln_w   = (const float*)d_in[11];
  const float* ln_b   = (const float*)d_in[12];
  const float* dec_w  = (const float*)d_in[13];
  const float* dec_b  = (const float*)d_in[14];
  float* out = (float*)d_out;

  float* ws = (float*)d_ws;
  size_t off = 0;
  float* coeff  = ws + off; off += (size_t)HH * NN * 4;        // w, 2Cc per mode
  float* Kbuf   = ws + off; off += (size_t)HH * CT;            // K[h][t]
  float* Ebuf   = ws + off; off += (size_t)HH * 64 * CT;       // E per h
  float* MVbuf  = ws + off; off += (size_t)HH * CT * KW;       // [M|V] per h
  float* states = ws + off; off += (size_t)BB * HH * NC * 64;  // chunk states
  float* hbuf   = ws + off; off += (size_t)BB * HH * LL;       // hidden
  float* gbuf   = ws + off; off += (size_t)BB * HH * LL;       // gelu(y)

  encoder_kernel<<<BB * (LL / 16), 256, 0, stream>>>(x, enc_w, enc_b, hbuf);

  for (int i = 0; i < NLAYER; ++i) {
    const float* ld  = log_dt + i * HH;
    const float* are = A_re + (size_t)i * HH * NN;
    const float* aim = A_im + (size_t)i * HH * NN;
    const float* cre = C_re + (size_t)i * HH * NN;
    const float* cim = C_im + (size_t)i * HH * NN;

    prep_coeff_ev<<<(HH * NN) / 256, 256, 0, stream>>>(ld, are, aim, cre, cim,
                                                       (float4*)coeff, Ebuf, MVbuf);
    prep_K<<<(HH * CT) / 256, 256, 0, stream>>>((const float4*)coeff, ld, are, aim, Kbuf);
    prep_Mfill<<<(HH * CT) / 256, 256, 0, stream>>>(Kbuf, Dv + i * HH, MVbuf);

    gemm_states<<<HH * 4, 256, 0, stream>>>(Ebuf, hbuf, states);
    combine_kernel<<<(BB * HH * NN) / 256, 256, 0, stream>>>((const float4*)coeff, states);
    gemm_y<<<HH * 8, 256, 0, stream>>>(MVbuf, hbuf, states, gbuf);

    wout_ln_kernel<<<BB * (LL / 16), 256, 0, stream>>>(gbuf, W_out + (size_t)i * HH * HH,
                                                       b_out + i * HH, hbuf,
                                                       ln_w + i * HH, ln_b + i * HH);
  }

  decoder_kernel<<<(BB * (LL / 16)) / 8, 256, 0, stream>>>(hbuf, dec_w, dec_b, out);
}